// Random_RNN_28037546508575
// MI455X (gfx1250) — compile-verified
//
#include <hip/hip_runtime.h>

// Problem constants (fixed by the reference file).
#define IN_F   256
#define N_ASS  4096
#define OUT_F  256
#define BATCH  128
#define ASS0   IN_F              // 256   : first associative neuron
#define OUT0   (IN_F + N_ASS)    // 4352  : first output neuron

typedef float v2f __attribute__((ext_vector_type(2)));
typedef float v8f __attribute__((ext_vector_type(8)));

// ---------------------------------------------------------------------------
// Zero a float buffer.
__global__ __launch_bounds__(256) void rnn_zero_f32(float* __restrict__ p, int n) {
    int i = blockIdx.x * blockDim.x + threadIdx.x;
    if (i < n) p[i] = 0.0f;
}

// Densify hop-1 edges: W1[in_dst-ASS0][in_src] += w   (W1 is [N_ASS x IN_F])
__global__ __launch_bounds__(256) void rnn_scatter_w1(const float* __restrict__ w,
                                                      const int* __restrict__ src,
                                                      const int* __restrict__ dst,
                                                      int e, float* __restrict__ W1) {
    int i = blockIdx.x * blockDim.x + threadIdx.x;
    if (i < e) {
        atomicAdd(&W1[(size_t)(dst[i] - ASS0) * IN_F + src[i]], w[i]);
    }
}

// Densify hop-2 edges that reach output neurons only:
// W2[ass_dst-OUT0][ass_src-ASS0] += w   (W2 is [OUT_F x N_ASS]).
// Edges into associative neurons never influence the read-out rows -> skip.
__global__ __launch_bounds__(256) void rnn_scatter_w2(const float* __restrict__ w,
                                                      const int* __restrict__ src,
                                                      const int* __restrict__ dst,
                                                      int e, float* __restrict__ W2) {
    int i = blockIdx.x * blockDim.x + threadIdx.x;
    if (i < e) {
        int d = dst[i];
        if (d >= OUT0) {
            atomicAdd(&W2[(size_t)(d - OUT0) * N_ASS + (src[i] - ASS0)], w[i]);
        }
    }
}

// ---------------------------------------------------------------------------
// Dense fp32 GEMM on V_WMMA_F32_16X16X4_F32.
//   C[M x N] = A[M x K] * B
//   BT=true : B supplied as [N x K] row-major (pre-transposed) -> both A and B
//             fragments are single contiguous 8B (v2f / b64) loads.
//   CT=true : store C transposed as [N x M] row-major.
// One wave computes one 16x16 tile: blockIdx.x = M-tile, wave-in-block = N-tile
// (8 waves -> N = 128). Waves in a block share A rows for L0 reuse.
// Fragment layouts per CDNA5 ISA 7.12.2:
//   A (16x4 fp32): lane l, vgpr v -> (M = l%16, K = v + 2*(l/16))
//   B (4x16 fp32): lane l, vgpr v -> (K = v + 2*(l/16), N = l%16)
//   C (16x16 f32): lane l, vgpr v -> (M = v + 8*(l/16), N = l%16)
template <int M, int N, int K, bool BT, bool CT>
__global__ __launch_bounds__(256) void rnn_wmma_gemm_f32(const float* __restrict__ A,
                                                         const float* __restrict__ B,
                                                         float* __restrict__ C) {
    const int lane = threadIdx.x & 31;
    const int wave = threadIdx.x >> 5;   // N-tile index
    const int tm   = blockIdx.x;         // M-tile index
    const int half = lane >> 4;          // 0: lanes 0-15, 1: lanes 16-31
    const int l16  = lane & 15;
    const int m    = tm * 16 + l16;      // A row held by this lane
    const int n    = wave * 16 + l16;    // B/C column held by this lane

    v8f acc = {};
    const float* __restrict__ Arow = A + (size_t)m * K;
    const float* __restrict__ Brow = BT ? (B + (size_t)n * K) : nullptr;

    for (int kb = 0; kb < K; kb += 4) {
        const int ka = kb + 2 * half;            // even -> 8B-aligned v2f loads
        v2f a = *(const v2f*)(Arow + ka);        // A(m, ka), A(m, ka+1)
        v2f b;
        if (BT) {
            b = *(const v2f*)(Brow + ka);        // B^T(n, ka..ka+1)
        } else {
            b.x = B[(size_t)ka * N + n];         // B(ka,   n)
            b.y = B[(size_t)(ka + 1) * N + n];   // B(ka+1, n)
        }
        // D = A x B + C, full fp32, 8 args: (neg_a, A, neg_b, B, c_mod, C, reuse_a, reuse_b)
        acc = __builtin_amdgcn_wmma_f32_16x16x4_f32(false, a, false, b,
                                                    (short)0, acc, false, false);
    }

#pragma unroll
    for (int v = 0; v < 8; ++v) {
        const int mr = tm * 16 + v + 8 * half;   // output row of this element
        if (CT) C[(size_t)n * M + mr] = acc[v];  // C stored as [N x M]
        else    C[(size_t)mr * N + n] = acc[v];  // C stored as [M x N]
    }
}

// ---------------------------------------------------------------------------
extern "C" void kernel_launch(void* const* d_in, const int* in_sizes, int n_in,
                              void* d_out, int out_size, void* d_ws, size_t ws_size,
                              hipStream_t stream) {
    const float* x   = (const float*)d_in[0];                 // [128 x 256]
    const float* wi  = (const float*)d_in[1];                 // [E_in]
    const float* wa  = (const float*)d_in[2];                 // [E_ass]
    const int* in_src  = (const int*)d_in[3];
    const int* in_dst  = (const int*)d_in[4];
    const int* ass_src = (const int*)d_in[5];
    const int* ass_dst = (const int*)d_in[6];
    const int e_in  = in_sizes[1];
    const int e_ass = in_sizes[2];

    // Workspace layout (fp32): W1 [4096x256] | W2 [256x4096] | ACC_T [128x4096]
    float* W1    = (float*)d_ws;
    float* W2    = W1 + (size_t)N_ASS * IN_F;
    float* ACC_T = W2 + (size_t)OUT_F * N_ASS;

    // 1) Zero the dense weight matrices (contiguous: W1 then W2).
    {
        const int n = N_ASS * IN_F + OUT_F * N_ASS;           // 2,097,152
        rnn_zero_f32<<<(n + 255) / 256, 256, 0, stream>>>(W1, n);
    }
    // 2) Scatter sparse edges into dense operators.
    rnn_scatter_w1<<<(e_in + 255) / 256, 256, 0, stream>>>(wi, in_src, in_dst, e_in, W1);
    rnn_scatter_w2<<<(e_ass + 255) / 256, 256, 0, stream>>>(wa, ass_src, ass_dst, e_ass, W2);

    // 3) Hop 1: ACC_T[128 x 4096] = (W1[4096 x 256] @ x.T).T
    //    x is [128 x 256] row-major -> BT gives contiguous b64 B loads;
    //    CT store produces the transposed layout GEMM2 wants.
    rnn_wmma_gemm_f32<N_ASS, BATCH, IN_F, true, true>
        <<<N_ASS / 16, 256, 0, stream>>>(W1, x, ACC_T);

    // 4) Hop 2 + transpose: out[128 x 256] = (W2[256 x 4096] @ ACC_T.T).T
    //    BT=true: dominant K=4096 loop uses only contiguous b64 loads.
    rnn_wmma_gemm_f32<OUT_F, BATCH, N_ASS, true, true>
        <<<OUT_F / 16, 256, 0, stream>>>(W2, ACC_T, (float*)d_out);

    (void)n_in; (void)out_size; (void)ws_size;
}